// LSSTransform_50629074485853
// MI455X (gfx1250) — compile-verified
//
#include <hip/hip_runtime.h>
#include <hip/hip_bf16.h>
#include <stddef.h>

// ---------------------------------------------------------------------------
// LSS (lift-splat) BEV transform for MI455X / gfx1250, wave32.
//   feat GEMM:   f32 WMMA 16x16x4 (full precision; kernel is mem/atomic bound)
//   weights:     pre-padded to 176 rows in workspace -> branch-free inner loop
//   splat:       LDS-staged softmax + geometry, L2 float atomics
// ---------------------------------------------------------------------------

typedef __attribute__((ext_vector_type(2))) float v2f;
typedef __attribute__((ext_vector_type(8))) float v8f;

#define NCAM   6
#define CIN    512
#define FH     16
#define FW     44
#define HW     (FH * FW)          // 704
#define DBINS  41
#define NCH    128
#define OREAL  (DBINS + NCH)      // 169
#define OPAD   176                // 11 * 16
#define NX     128
#define NY     128

// ---------------------------------------------------------------------------
// Zero the output (harness poisons d_out; we accumulate with atomics).
// ---------------------------------------------------------------------------
__global__ void zero_out_kernel(float* __restrict__ out, int n) {
    int i = blockIdx.x * blockDim.x + threadIdx.x;
    if (i < n) out[i] = 0.0f;
}

// ---------------------------------------------------------------------------
// Pad W [169,512] -> Wpad [176,512] (zero rows) and bias -> biaspad [176].
// One-shot; removes all OOB guards from the GEMM inner loop.
// ---------------------------------------------------------------------------
__global__ void pad_w_kernel(const float* __restrict__ w,
                             const float* __restrict__ bias,
                             float* __restrict__ wpad,
                             float* __restrict__ biaspad) {
    int i = blockIdx.x * blockDim.x + threadIdx.x;
    if (i < OPAD * CIN) {
        int o = i / CIN;                       // same flat index in both
        wpad[i] = (o < OREAL) ? w[i] : 0.0f;
    }
    if (i < OPAD) biaspad[i] = (i < OREAL) ? bias[i] : 0.0f;
}

// ---------------------------------------------------------------------------
// Per-camera matrix setup: combine = rots @ inv(intrins), inv_post, trans,
// post_trans  ->  cam[n*24 .. n*24+23]
// ---------------------------------------------------------------------------
__device__ inline void inv3x3(const float* m, float* inv) {
    float a = m[0], b = m[1], c = m[2];
    float d = m[3], e = m[4], f = m[5];
    float g = m[6], h = m[7], i = m[8];
    float A = e * i - f * h;
    float B = -(d * i - f * g);
    float C = d * h - e * g;
    float det = a * A + b * B + c * C;
    float id = 1.0f / det;
    inv[0] = A * id;
    inv[1] = -(b * i - c * h) * id;
    inv[2] = (b * f - c * e) * id;
    inv[3] = B * id;
    inv[4] = (a * i - c * g) * id;
    inv[5] = -(a * f - c * d) * id;
    inv[6] = C * id;
    inv[7] = -(a * h - b * g) * id;
    inv[8] = (a * e - b * d) * id;
}

__global__ void cam_setup_kernel(const float* __restrict__ rots,
                                 const float* __restrict__ trans,
                                 const float* __restrict__ intrins,
                                 const float* __restrict__ post_rots,
                                 const float* __restrict__ post_trans,
                                 float* __restrict__ cam) {
    int n = threadIdx.x;
    if (n >= NCAM) return;
    float invK[9], invP[9];
    inv3x3(intrins + n * 9, invK);
    inv3x3(post_rots + n * 9, invP);
    const float* R = rots + n * 9;
    float* o = cam + n * 24;
    for (int r = 0; r < 3; ++r)
        for (int c = 0; c < 3; ++c) {
            float s = 0.0f;
            for (int k = 0; k < 3; ++k) s += R[r * 3 + k] * invK[k * 3 + c];
            o[r * 3 + c] = s;                 // combine
        }
    for (int j = 0; j < 9; ++j) o[9 + j] = invP[j];
    o[18] = trans[n * 3 + 0];
    o[19] = trans[n * 3 + 1];
    o[20] = trans[n * 3 + 2];
    o[21] = post_trans[n * 3 + 0];
    o[22] = post_trans[n * 3 + 1];
    o[23] = post_trans[n * 3 + 2];
}

// ---------------------------------------------------------------------------
// GEMM: feat[n, o, p] = sum_c Wpad[o, c] * x[n, c, p] + biaspad[o]
// One wave computes a 16(o) x 64(p) tile with 4 f32 accumulators (A reused).
// f32 WMMA fragment layouts (ISA 7.12.2, 32-bit, wave32):
//   A 16x4 : lane<16 -> M=lane,   {v.x,v.y} = K{0,1};  lane>=16 -> K{2,3}
//   B 4x16 : lane<16 -> N=lane,   {v.x,v.y} = K{0,1};  lane>=16 -> K{2,3}
//   C 16x16: VGPR j -> M = j + 8*(lane>>4), N = lane&15
// Inner loop is branch-free: 1x b64 (A) + 8x b32 (B) loads -> 4 WMMAs.
// ---------------------------------------------------------------------------
__global__ __launch_bounds__(32) void gemm_feat_kernel(
    const float* __restrict__ x,     // [NCAM, CIN, HW]
    const float* __restrict__ wpad,  // [OPAD, CIN], padded rows are zero
    const float* __restrict__ biaspad, // [OPAD]
    float* __restrict__ feat)        // [NCAM, OPAD, HW]
{
    const int n    = blockIdx.x;     // camera
    const int ot   = blockIdx.y;     // o tile (0..10)
    const int pt   = blockIdx.z;     // p super-tile (0..10), 64 wide
    const int lane = threadIdx.x;
    const int lr   = lane & 15;
    const int hi   = lane >> 4;

    const float* xb   = x + (size_t)n * CIN * HW;
    const float* wrow = wpad + (size_t)(ot * 16 + lr) * CIN;  // 2KB-aligned row
    const int p0 = pt * 64 + lr;

    v8f acc0 = {}; v8f acc1 = {}; v8f acc2 = {}; v8f acc3 = {};

    for (int k0 = 0; k0 < CIN; k0 += 4) {
        const int ka = k0 + 2 * hi;                  // even -> 8B aligned
        v2f a = *(const v2f*)(wrow + ka);            // single b64 load
        const float* xk0 = xb + (size_t)ka * HW;
        const float* xk1 = xk0 + HW;
        v2f b0; b0.x = xk0[p0];      b0.y = xk1[p0];
        v2f b1; b1.x = xk0[p0 + 16]; b1.y = xk1[p0 + 16];
        v2f b2; b2.x = xk0[p0 + 32]; b2.y = xk1[p0 + 32];
        v2f b3; b3.x = xk0[p0 + 48]; b3.y = xk1[p0 + 48];
        acc0 = __builtin_amdgcn_wmma_f32_16x16x4_f32(false, a, false, b0,
                                                     (short)0, acc0, false, false);
        acc1 = __builtin_amdgcn_wmma_f32_16x16x4_f32(false, a, false, b1,
                                                     (short)0, acc1, false, false);
        acc2 = __builtin_amdgcn_wmma_f32_16x16x4_f32(false, a, false, b2,
                                                     (short)0, acc2, false, false);
        acc3 = __builtin_amdgcn_wmma_f32_16x16x4_f32(false, a, false, b3,
                                                     (short)0, acc3, false, false);
    }

    float* fb = feat + (size_t)n * OPAD * HW;
    for (int j = 0; j < 8; ++j) {
        const int o  = ot * 16 + j + 8 * hi;
        const float bv = biaspad[o];
        float* row = fb + (size_t)o * HW + pt * 64 + lr;
        row[0]  = acc0[j] + bv;
        row[16] = acc1[j] + bv;
        row[32] = acc2[j] + bv;
        row[48] = acc3[j] + bv;
    }
}

// ---------------------------------------------------------------------------
// Splat: one 128-thread block per (camera, pixel).
//   - stage feat[0..168] in LDS
//   - softmax over first 41 (depth)
//   - threads 0..40 project frustum point d -> voxel cell (or -1)
//   - thread c (0..127) scatters depth[d]*cfeat[c] into out[c, iy, ix]
// ---------------------------------------------------------------------------
__global__ __launch_bounds__(128) void splat_kernel(
    const float* __restrict__ feat,  // [NCAM, OPAD, HW]
    const float* __restrict__ cam,   // [NCAM, 24]
    float* __restrict__ out)         // [NCH, NY, NX]
{
    __shared__ float s_feat[OREAL];
    __shared__ float s_depth[DBINS];
    __shared__ int   s_cell[DBINS];

    const int pix = blockIdx.x;          // 0 .. NCAM*HW-1
    const int n   = pix / HW;
    const int p   = pix % HW;
    const int h   = p / FW;
    const int w   = p % FW;
    const int tid = threadIdx.x;

    const float* fb = feat + (size_t)n * OPAD * HW + p;
    for (int i = tid; i < OREAL; i += 128) s_feat[i] = fb[(size_t)i * HW];
    __syncthreads();

    if (tid == 0) {
        float m = s_feat[0];
        for (int d = 1; d < DBINS; ++d) m = fmaxf(m, s_feat[d]);
        float sum = 0.0f;
        for (int d = 0; d < DBINS; ++d) {
            float e = __expf(s_feat[d] - m);
            s_depth[d] = e;
            sum += e;
        }
        float inv = 1.0f / sum;
        for (int d = 0; d < DBINS; ++d) s_depth[d] *= inv;
    }

    if (tid < DBINS) {
        const float* cm = cam + n * 24;
        // frustum point (same f32 arithmetic as the reference linspace/arange)
        const float fx = (float)w * (703.0f / 43.0f);
        const float fy = (float)h * (255.0f / 15.0f);
        const float fd = 4.0f + (float)tid;
        // undo post-transform
        float px = fx - cm[21], py = fy - cm[22], pz = fd - cm[23];
        float qx = cm[9]  * px + cm[10] * py + cm[11] * pz;
        float qy = cm[12] * px + cm[13] * py + cm[14] * pz;
        float qz = cm[15] * px + cm[16] * py + cm[17] * pz;
        // [x*z, y*z, z]
        float rx = qx * qz, ry = qy * qz, rz = qz;
        // combine @ r + trans
        float gx = cm[0] * rx + cm[1] * ry + cm[2] * rz + cm[18];
        float gy = cm[3] * rx + cm[4] * ry + cm[5] * rz + cm[19];
        float gz = cm[6] * rx + cm[7] * ry + cm[8] * rz + cm[20];
        // voxelize: (geom - (BX - DX/2)) / DX, truncation-toward-zero cast
        const float bx0 = -50.8f, by0 = -50.8f, bz0 = 0.0f;
        const float dx0 = 0.8f,   dy0 = 0.8f,   dz0 = 20.0f;
        const float ox = bx0 - dx0 * 0.5f;
        const float oy = by0 - dy0 * 0.5f;
        const float oz = bz0 - dz0 * 0.5f;
        int ix = (int)((gx - ox) / dx0);
        int iy = (int)((gy - oy) / dy0);
        int iz = (int)((gz - oz) / dz0);
        bool valid = (ix >= 0) && (ix < NX) && (iy >= 0) && (iy < NY) && (iz == 0);
        s_cell[tid] = valid ? (iy * NX + ix) : -1;
    }
    __syncthreads();

    const float cf = s_feat[DBINS + tid];           // channel feature
    float* plane = out + (size_t)tid * (NY * NX);   // channel plane
    for (int d = 0; d < DBINS; ++d) {
        int cell = s_cell[d];
        if (cell >= 0) atomicAdd(plane + cell, s_depth[d] * cf);
    }
}

// ---------------------------------------------------------------------------
// Launch
// ---------------------------------------------------------------------------
extern "C" void kernel_launch(void* const* d_in, const int* in_sizes, int n_in,
                              void* d_out, int out_size, void* d_ws, size_t ws_size,
                              hipStream_t stream) {
    const float* x          = (const float*)d_in[0]; // [1,6,512,16,44]
    const float* rots       = (const float*)d_in[1]; // [1,6,3,3]
    const float* trans      = (const float*)d_in[2]; // [1,6,3]
    const float* intrins    = (const float*)d_in[3]; // [1,6,3,3]
    const float* post_rots  = (const float*)d_in[4]; // [1,6,3,3]
    const float* post_trans = (const float*)d_in[5]; // [1,6,3]
    const float* w_depth    = (const float*)d_in[6]; // [169,512]
    const float* b_depth    = (const float*)d_in[7]; // [169]
    float* out = (float*)d_out;                      // [1,128,128,128]

    // workspace layout (floats)
    float* feat_ws = (float*)d_ws;                           // NCAM*OPAD*HW
    float* cam_ws  = feat_ws + (size_t)NCAM * OPAD * HW;     // NCAM*24
    float* wpad_ws = cam_ws + NCAM * 24;                     // OPAD*CIN
    float* bias_ws = wpad_ws + (size_t)OPAD * CIN;           // OPAD

    zero_out_kernel<<<(out_size + 255) / 256, 256, 0, stream>>>(out, out_size);

    pad_w_kernel<<<(OPAD * CIN + 255) / 256, 256, 0, stream>>>(
        w_depth, b_depth, wpad_ws, bias_ws);

    cam_setup_kernel<<<1, 32, 0, stream>>>(rots, trans, intrins, post_rots,
                                           post_trans, cam_ws);

    dim3 ggrid(NCAM, OPAD / 16, HW / 64);   // 6 x 11 x 11 waves
    gemm_feat_kernel<<<ggrid, 32, 0, stream>>>(x, wpad_ws, bias_ws, feat_ws);

    splat_kernel<<<NCAM * HW, 128, 0, stream>>>(feat_ws, cam_ws, out);
}